// GNN_24369644437586
// MI455X (gfx1250) — compile-verified
//
#include <hip/hip_runtime.h>

#define INPT  64
#define HID   128
#define NNODE 10000
#define KNB   16
#define BATCH 8

// LDS row strides (floats), padded so row-to-row bank offset is 4 (or 8) of
// the 64 LDS banks, while keeping every row 16-byte aligned for float4 ops.
#define STR_A64  68    // 64  + 4 : A tile, K=64
#define STR_A128 132   // 128 + 4 : A tile, K=128
#define STR_AU   260   // 256 + 4 : A tile, K=256 (aggr || self)
#define STR_W    136   // 128 + 8 : staged W1 rows

typedef __attribute__((ext_vector_type(2))) float v2f;
typedef __attribute__((ext_vector_type(8))) float v8f;
typedef __attribute__((ext_vector_type(4))) int   v4i;

// ---- async global->LDS staging (CDNA5 ASYNCcnt path), with sync fallback ----
#if defined(__has_builtin)
#  if __has_builtin(__builtin_amdgcn_global_load_async_to_lds_b128)
#    define HAVE_ASYNC_B128 1
#  endif
#  if __has_builtin(__builtin_amdgcn_s_wait_asynccnt)
#    define HAVE_WAIT_ASYNC 1
#  endif
#endif
#ifndef HAVE_ASYNC_B128
#  define HAVE_ASYNC_B128 0
#endif
#ifndef HAVE_WAIT_ASYNC
#  define HAVE_WAIT_ASYNC 0
#endif

// Builtin signature (from hipcc diagnostic): param 0 is AS1 pointer to v4i,
// param 1 is the LDS-side pointer, then imm offset + imm cpol.
typedef __attribute__((address_space(1))) v4i gas_v4i;
typedef __attribute__((address_space(3))) v4i las_v4i;

__device__ __forceinline__ void copy16_to_lds(float* l, const float* g) {
#if HAVE_ASYNC_B128
  // On CDNA5 the generic->LDS aperture conversion is addr[31:0] truncation.
  __builtin_amdgcn_global_load_async_to_lds_b128(
      (gas_v4i*)(unsigned long long)(uintptr_t)g,
      (las_v4i*)(unsigned int)(uintptr_t)l, 0, 0);
#else
  *(float4*)l = *(const float4*)g;
#endif
}

__device__ __forceinline__ void lds_copy_fence() {
#if HAVE_ASYNC_B128
#  if HAVE_WAIT_ASYNC
  __builtin_amdgcn_s_wait_asynccnt(0);
#  else
  asm volatile("s_wait_asynccnt 0" ::: "memory");
#  endif
#endif
}

// ---------------------------------------------------------------------------
// Y[r, 0:128] = relu(X[r, 0:64] @ W[64,128] + b)
// Block: 256 threads (8 waves). Block tile: 32 rows x 128 cols.
// Wave w: two 16x16 WMMA tiles (rows 0-15 / 16-31) at cols [16w,16w+16),
// sharing each B fragment. A tile + W + bias staged in LDS (conflict-padded).
// ---------------------------------------------------------------------------
__global__ __launch_bounds__(256) void gemm64_relu(const float* __restrict__ X,
                                                   const float* __restrict__ W,
                                                   const float* __restrict__ bias,
                                                   float* __restrict__ Y) {
  __shared__ float ldsW[INPT * STR_W];    // 64 rows x 136  (~34 KB)
  __shared__ float ldsA[32 * STR_A64];    // 32 rows x 68   (~8.7 KB)
  __shared__ float ldsB[HID];

  const long row0 = (long)blockIdx.x * 32;

  // stage W (64x128) row-wise with padding: 2048 float4 tasks (async path)
#pragma unroll
  for (int i = 0; i < 8; ++i) {
    const int id  = threadIdx.x + 256 * i;
    const int row = id >> 5, c4 = (id & 31) << 2;
    copy16_to_lds(&ldsW[row * STR_W + c4], &W[row * HID + c4]);
  }
  // stage A (32x64): 512 float4 tasks
#pragma unroll
  for (int i = 0; i < 2; ++i) {
    const int id  = threadIdx.x + 256 * i;
    const int row = id >> 4, c4 = (id & 15) << 2;
    copy16_to_lds(&ldsA[row * STR_A64 + c4], &X[(row0 + row) * INPT + c4]);
  }
  if (threadIdx.x < HID) ldsB[threadIdx.x] = bias[threadIdx.x];
  lds_copy_fence();
  __syncthreads();

  const int wave = threadIdx.x >> 5;
  const int lane = threadIdx.x & 31;
  const int half = lane >> 4;
  const int mr   = lane & 15;
  const int col0 = wave * 16;

  v8f acc0 = {}, acc1 = {};
#pragma unroll
  for (int s = 0; s < INPT / 4; ++s) {
    const int k = 4 * s + 2 * half;
    v2f Bf; Bf.x = ldsW[(k)     * STR_W + col0 + mr];
            Bf.y = ldsW[(k + 1) * STR_W + col0 + mr];
    v2f A0; A0.x = ldsA[(mr)      * STR_A64 + k];
            A0.y = ldsA[(mr)      * STR_A64 + k + 1];
    v2f A1; A1.x = ldsA[(16 + mr) * STR_A64 + k];
            A1.y = ldsA[(16 + mr) * STR_A64 + k + 1];
    acc0 = __builtin_amdgcn_wmma_f32_16x16x4_f32(false, A0, false, Bf,
                                                 (short)0, acc0, false, false);
    acc1 = __builtin_amdgcn_wmma_f32_16x16x4_f32(false, A1, false, Bf,
                                                 (short)0, acc1, false, false);
  }

  const float bv = ldsB[col0 + mr];
#pragma unroll
  for (int v = 0; v < 8; ++v) {
    float x0 = acc0[v] + bv;
    float x1 = acc1[v] + bv;
    Y[(row0 + v + 8 * half)      * HID + col0 + mr] = x0 > 0.f ? x0 : 0.f;
    Y[(row0 + 16 + v + 8 * half) * HID + col0 + mr] = x1 > 0.f ? x1 : 0.f;
  }
}

// ---------------------------------------------------------------------------
// Y[r, 0:128] = relu(X[r, 0:128] @ W[128,128] + b)
// A tile staged in LDS; W (64 KB, L2-resident, shared by all blocks) from global.
// ---------------------------------------------------------------------------
__global__ __launch_bounds__(256) void gemm128_relu(const float* __restrict__ X,
                                                    const float* __restrict__ W,
                                                    const float* __restrict__ bias,
                                                    float* __restrict__ Y) {
  __shared__ float ldsA[32 * STR_A128];   // 32 rows x 132 (~16.9 KB)

  const long row0 = (long)blockIdx.x * 32;

  // stage A (32x128): 1024 float4 tasks (async path)
#pragma unroll
  for (int i = 0; i < 4; ++i) {
    const int id  = threadIdx.x + 256 * i;
    const int row = id >> 5, c4 = (id & 31) << 2;
    copy16_to_lds(&ldsA[row * STR_A128 + c4], &X[(row0 + row) * HID + c4]);
  }
  lds_copy_fence();
  __syncthreads();

  const int wave = threadIdx.x >> 5;
  const int lane = threadIdx.x & 31;
  const int half = lane >> 4;
  const int mr   = lane & 15;
  const int col0 = wave * 16;

  v8f acc0 = {}, acc1 = {};
#pragma unroll 8
  for (int s = 0; s < HID / 4; ++s) {
    const int k = 4 * s + 2 * half;
    v2f Bf; Bf.x = W[(k)     * HID + col0 + mr];
            Bf.y = W[(k + 1) * HID + col0 + mr];
    v2f A0; A0.x = ldsA[(mr)      * STR_A128 + k];
            A0.y = ldsA[(mr)      * STR_A128 + k + 1];
    v2f A1; A1.x = ldsA[(16 + mr) * STR_A128 + k];
            A1.y = ldsA[(16 + mr) * STR_A128 + k + 1];
    acc0 = __builtin_amdgcn_wmma_f32_16x16x4_f32(false, A0, false, Bf,
                                                 (short)0, acc0, false, false);
    acc1 = __builtin_amdgcn_wmma_f32_16x16x4_f32(false, A1, false, Bf,
                                                 (short)0, acc1, false, false);
  }

  const float bv = bias[col0 + mr];
#pragma unroll
  for (int v = 0; v < 8; ++v) {
    float x0 = acc0[v] + bv;
    float x1 = acc1[v] + bv;
    Y[(row0 + v + 8 * half)      * HID + col0 + mr] = x0 > 0.f ? x0 : 0.f;
    Y[(row0 + 16 + v + 8 * half) * HID + col0 + mr] = x1 > 0.f ? x1 : 0.f;
  }
}

// ---------------------------------------------------------------------------
// Fused aggregate + update:
//   aggr[b,n,:] = sum_k se[b, sidx[n,k], :] + sum_k ie[b, iidx[n,k], :]
//   Y[b,n,:]    = relu(concat(aggr, ie[b,n,:]) @ Wu[256,128] + bu)
// 32 rows per block; b/n resolved per row (tile may straddle a batch edge).
// Phase 1 builds the 32x256 A tile in LDS (gathers are L2-served: both embed
// buffers total 82 MB < 192 MB L2). Self-row copy uses the async LDS path.
// Phase 2: 64 K-steps x 2 WMMA per wave.
// ---------------------------------------------------------------------------
__global__ __launch_bounds__(256) void aggregate_update(
    const float* __restrict__ se, const float* __restrict__ ie,
    const int* __restrict__ sidx, const int* __restrict__ iidx,
    const float* __restrict__ Wu, const float* __restrict__ bu,
    float* __restrict__ Y) {
  __shared__ float ldsA[32 * STR_AU];     // 32 rows x 260 (~33 KB)

  const long gr0 = (long)blockIdx.x * 32;

  // ---- phase 1: gather-sum + self-row copy (512 row x col-group tasks) ----
#pragma unroll
  for (int i = 0; i < 2; ++i) {
    const int id = threadIdx.x + 256 * i;
    const int r  = id >> 4;              // row 0..31
    const int cg = (id & 15) * 8;        // 8-float column group
    const long gr = gr0 + r;
    const int  b  = (int)(gr / NNODE);
    const int  n  = (int)(gr % NNODE);
    const float* seB = se + (long)b * NNODE * HID;
    const float* ieB = ie + (long)b * NNODE * HID;

    // self row (concat second half) via async LDS path, overlapped with gather
    copy16_to_lds(&ldsA[r * STR_AU + HID + cg],     ieB + (long)n * HID + cg);
    copy16_to_lds(&ldsA[r * STR_AU + HID + cg + 4], ieB + (long)n * HID + cg + 4);

    float a[8];
#pragma unroll
    for (int j = 0; j < 8; ++j) a[j] = 0.f;

#pragma unroll
    for (int k = 0; k < KNB; ++k) {
      const int sn = sidx[n * KNB + k];
      const float4* p = (const float4*)(seB + (long)sn * HID + cg);
      const float4 p0 = p[0], p1 = p[1];
      const int in_ = iidx[n * KNB + k];
      const float4* q = (const float4*)(ieB + (long)in_ * HID + cg);
      const float4 q0 = q[0], q1 = q[1];
      a[0] += p0.x + q0.x;  a[1] += p0.y + q0.y;
      a[2] += p0.z + q0.z;  a[3] += p0.w + q0.w;
      a[4] += p1.x + q1.x;  a[5] += p1.y + q1.y;
      a[6] += p1.z + q1.z;  a[7] += p1.w + q1.w;
    }

    *(float4*)&ldsA[r * STR_AU + cg]     = make_float4(a[0], a[1], a[2], a[3]);
    *(float4*)&ldsA[r * STR_AU + cg + 4] = make_float4(a[4], a[5], a[6], a[7]);
  }
  lds_copy_fence();
  __syncthreads();

  // ---- phase 2: two 16x16 WMMA tiles per wave, K = 256 ----
  const int wave = threadIdx.x >> 5;
  const int lane = threadIdx.x & 31;
  const int half = lane >> 4;
  const int mr   = lane & 15;
  const int col0 = wave * 16;

  v8f acc0 = {}, acc1 = {};
#pragma unroll 8
  for (int s = 0; s < (2 * HID) / 4; ++s) {
    const int k = 4 * s + 2 * half;
    v2f Bf; Bf.x = Wu[(k)     * HID + col0 + mr];
            Bf.y = Wu[(k + 1) * HID + col0 + mr];
    v2f A0; A0.x = ldsA[(mr)      * STR_AU + k];
            A0.y = ldsA[(mr)      * STR_AU + k + 1];
    v2f A1; A1.x = ldsA[(16 + mr) * STR_AU + k];
            A1.y = ldsA[(16 + mr) * STR_AU + k + 1];
    acc0 = __builtin_amdgcn_wmma_f32_16x16x4_f32(false, A0, false, Bf,
                                                 (short)0, acc0, false, false);
    acc1 = __builtin_amdgcn_wmma_f32_16x16x4_f32(false, A1, false, Bf,
                                                 (short)0, acc1, false, false);
  }

  const float bv = bu[col0 + mr];
#pragma unroll
  for (int v = 0; v < 8; ++v) {
    float x0 = acc0[v] + bv;
    float x1 = acc1[v] + bv;
    Y[(gr0 + v + 8 * half)      * HID + col0 + mr] = x0 > 0.f ? x0 : 0.f;
    Y[(gr0 + 16 + v + 8 * half) * HID + col0 + mr] = x1 > 0.f ? x1 : 0.f;
  }
}

// ---------------------------------------------------------------------------
extern "C" void kernel_launch(void* const* d_in, const int* in_sizes, int n_in,
                              void* d_out, int out_size, void* d_ws, size_t ws_size,
                              hipStream_t stream) {
  const float* state     = (const float*)d_in[0];
  const float* internal_ = (const float*)d_in[1];
  const int*   sidx      = (const int*)d_in[2];
  const int*   iidx      = (const int*)d_in[3];
  const float* W1  = (const float*)d_in[4];
  const float* b1  = (const float*)d_in[5];
  const float* W2  = (const float*)d_in[6];
  const float* b2  = (const float*)d_in[7];
  const float* Wu1 = (const float*)d_in[8];
  const float* bu1 = (const float*)d_in[9];
  const float* Wu2 = (const float*)d_in[10];
  const float* bu2 = (const float*)d_in[11];

  const long S = (long)BATCH * NNODE * HID;    // 10.24 M floats per tensor
  float* hu1 = (float*)d_out;
  float* hu2 = (float*)d_out + S;

  float* bufA = (float*)d_ws;          // se  -> later reused for ie2
  float* bufB = bufA + S;              // ie  -> later reused for se2

  const int RT = (BATCH * NNODE) / 32; // 2500 row tiles of 32
  dim3 grid(RT), block(256);

  // layer-1 embeddings
  gemm64_relu<<<grid, block, 0, stream>>>(state,     W1, b1, bufA);  // se
  gemm64_relu<<<grid, block, 0, stream>>>(internal_, W1, b1, bufB);  // ie
  // hu1 = relu(concat(aggr(se,ie), ie) @ Wu1 + bu1)
  aggregate_update<<<grid, block, 0, stream>>>(bufA, bufB, sidx, iidx,
                                               Wu1, bu1, hu1);
  // layer-2 embeddings (ping-pong the workspace buffers)
  gemm128_relu<<<grid, block, 0, stream>>>(bufA, W2, b2, bufB);      // se2 (over ie)
  gemm128_relu<<<grid, block, 0, stream>>>(hu1,  W2, b2, bufA);      // ie2 (over se)
  // hu2 = relu(concat(aggr(se2,ie2), ie2) @ Wu2 + bu2)
  aggregate_update<<<grid, block, 0, stream>>>(bufB, bufA, sidx, iidx,
                                               Wu2, bu2, hu2);
}